// LengthRegulator_59030030516336
// MI455X (gfx1250) — compile-verified
//
#include <hip/hip_runtime.h>
#include <stdint.h>

// Problem constants (fixed by the harness): B=16, N=256, D=512, T=2048
#define B_ 16
#define N_ 256
#define D_ 512
#define T_ 2048

// ---------------------------------------------------------------------------
// Hot kernel (placed first so the disasm snippet shows it):
// one 128-thread block per (batch, token). Stage the 2KB phoneme row into
// LDS via the CDNA5 async DMA path (ASYNCcnt-tracked), then replicate it to
// the token's [start, start+dur) contiguous output frames with async stores
// straight from LDS. Each lane owns a disjoint 16B chunk that it loaded
// itself, so a per-wave s_wait_asynccnt is the only sync required.
// ---------------------------------------------------------------------------
__global__ __launch_bounds__(128) void lr_expand(const float* __restrict__ phon,
                                                 const int* __restrict__ dur,
                                                 const int* __restrict__ starts,
                                                 float* __restrict__ out) {
    const int token = blockIdx.x;          // token = b*N + n (uniform)
    const int d     = dur[token];          // uniform -> scalar load
    if (d == 0) return;                    // uniform across block
    const int b     = token >> 8;          // N_ == 256
    const int start = starts[token];
    const int tid   = threadIdx.x;         // 128 threads * 16B = 2KB row

    __shared__ float row[D_];
    // Low 32 bits of the flat address of an LDS pointer == byte offset into
    // the workgroup's LDS allocation (aperture bits live in [63:32]).
    unsigned lds = (unsigned)(uintptr_t)(&row[0]) + (unsigned)tid * 16u;

    unsigned long long src =
        (unsigned long long)(uintptr_t)(phon + (size_t)token * D_) +
        (unsigned long long)tid * 16u;

    // Memory -> LDS, 16B per lane, tracked by ASYNCcnt.
    asm volatile("global_load_async_to_lds_b128 %0, %1, off"
                 :: "v"(lds), "v"(src)
                 : "memory");
    asm volatile("s_wait_asynccnt 0" ::: "memory");

    unsigned long long dst =
        (unsigned long long)(uintptr_t)(out + ((size_t)b * T_ + (size_t)start) * D_) +
        (unsigned long long)tid * 16u;

    // LDS -> memory, one fully-coalesced 2KB frame per iteration; up to 63
    // stores in flight per wave (hardware stalls issue at the counter limit).
    // NOTE: cannot use the instruction offset: field to step frames — for
    // async LDS ops INST_OFFSET is added to BOTH the global and LDS address,
    // which would walk the LDS read pointer out of the row. Step dst in VALU.
#pragma unroll 4
    for (int f = 0; f < d; ++f) {
        asm volatile("global_store_async_from_lds_b128 %0, %1, off"
                     :: "v"(dst), "v"(lds)
                     : "memory");
        dst += (unsigned long long)(D_ * 4);
    }
    asm volatile("s_wait_asynccnt 0" ::: "memory");
}

// ---------------------------------------------------------------------------
// Kernel 2: per-batch exclusive prefix sum of durations -> start frame of
// each token. 16 blocks x 256 threads, Hillis-Steele scan in LDS. Tiny.
// ---------------------------------------------------------------------------
__global__ __launch_bounds__(N_) void lr_scan(const int* __restrict__ dur,
                                              int* __restrict__ starts) {
    __shared__ int s[N_];
    const int t    = threadIdx.x;
    const int base = blockIdx.x * N_;
    const int d    = dur[base + t];
    s[t] = d;
    __syncthreads();
#pragma unroll
    for (int off = 1; off < N_; off <<= 1) {
        int v = (t >= off) ? s[t - off] : 0;
        __syncthreads();
        s[t] += v;
        __syncthreads();
    }
    starts[base + t] = s[t] - d;  // exclusive prefix = first frame of token
}

// ---------------------------------------------------------------------------
// Host-side launcher
// ---------------------------------------------------------------------------
extern "C" void kernel_launch(void* const* d_in, const int* in_sizes, int n_in,
                              void* d_out, int out_size, void* d_ws, size_t ws_size,
                              hipStream_t stream) {
    const float* phon = (const float*)d_in[0];  // (B, N, D) float32
    const int*   dur  = (const int*)d_in[1];    // (B, N)    int32
    // d_in[2] = total_len scalar (== T_, compile-time constant here)
    float* out    = (float*)d_out;              // (B, T, D) float32
    int*   starts = (int*)d_ws;                 // B*N ints = 16KB scratch

    lr_scan<<<B_, N_, 0, stream>>>(dur, starts);
    lr_expand<<<B_ * N_, 128, 0, stream>>>(phon, dur, starts, out);

    (void)in_sizes; (void)n_in; (void)out_size; (void)ws_size;
}